// train_model_27925877358676
// MI455X (gfx1250) — compile-verified
//
#include <hip/hip_runtime.h>
#include <hip/hip_bf16.h>

// ---------------------------------------------------------------------------
// CDNA5 (gfx1250, wave32) implementation.
// GEMMs run on V_WMMA_F32_16X16X4_F32 (native fp32 matrix pipe), staged
// through LDS with coalesced b128 global loads; full fp32 fidelity.
// ---------------------------------------------------------------------------

typedef __attribute__((ext_vector_type(2))) float v2f;
typedef __attribute__((ext_vector_type(8))) float v8f;

#define EPS 1e-12f
#define KB   16          // k-slice per LDS stage
#define APAD 2           // LDS row padding (bank-conflict avoidance)

// --------------------------- wave32 reductions -----------------------------
__device__ __forceinline__ float wave_sum(float v) {
    for (int off = 16; off > 0; off >>= 1) v += __shfl_xor(v, off, 32);
    return v;
}
__device__ __forceinline__ float wave_max(float v) {
    for (int off = 16; off > 0; off >>= 1) v = fmaxf(v, __shfl_xor(v, off, 32));
    return v;
}

// ===========================================================================
// Tiled WMMA f32 GEMM.
//   TRANSB=true : C = A[M,K] @ W[N,K]^T (+ A2 @ W2^T if DUAL) (+bias)(relu?)
//   TRANSB=false: C = A[M,K] @ B[K,N]                         (+bias)(relu?)
// Block: 256 threads = 8 waves. Macro tile: 128 (M) x 64 (N).
// Wave w computes rows [m0+16w, m0+16w+16) x 64 cols -> 4 WMMA accumulators.
// LDS staging per KB=16 k-slice:
//   As: 128 x 16 (padded stride 18)            9216 B
//   Bs: TRANSB ? 64 x 16 (stride 18) : 16 x 64 (stride 66)   <= 4608 B
// Fragment layout for V_WMMA_F32_16X16X4_F32 (A 16x4 / B 4x16, 2 VGPR each):
//   lane = half*16 + r holds K = {2*half, 2*half+1}; row/col index = r.
// C 16x16 (8 VGPR): row = j + 8*half, col = r.
// Requires: N % 64 == 0, K % 16 == 0; M guarded.
// ===========================================================================
template <bool TRANSB, bool DUAL, bool RELU>
__global__ __launch_bounds__(256) void wmma_gemm_f32_tiled(
    const float* __restrict__ A, int lda,
    const float* __restrict__ B, int ldb,
    const float* __restrict__ A2,
    const float* __restrict__ B2,
    const float* __restrict__ bias,
    float* __restrict__ C, int ldc,
    int M, int K)
{
    __shared__ float As[128 * (KB + APAD)];
    __shared__ float Bs[64 * (KB + APAD)];   // NN view uses 16 x (64+APAD) <= same size

    const int tid  = threadIdx.x;
    const int wave = tid >> 5;
    const int lane = tid & 31;
    const int half = lane >> 4;
    const int r    = lane & 15;
    const int m0   = blockIdx.x * 128;
    const int n0   = blockIdx.y * 64;

    v8f acc[4];
#pragma unroll
    for (int t = 0; t < 4; ++t) acc[t] = (v8f){0.f,0.f,0.f,0.f,0.f,0.f,0.f,0.f};

    const int npass = DUAL ? 2 : 1;
    for (int pass = 0; pass < npass; ++pass) {
        const float* __restrict__ Ap = (DUAL && pass) ? A2 : A;
        const float* __restrict__ Bp = (DUAL && pass) ? B2 : B;

        for (int k0 = 0; k0 < K; k0 += KB) {
            __syncthreads();   // protect LDS from previous slice's readers

            // ---- stage A tile: 128 x 16 = 512 float4, 2 per thread ----
#pragma unroll
            for (int i = tid; i < 512; i += 256) {
                const int row = i >> 2;
                const int col = (i & 3) * 4;
                int gr = m0 + row; if (gr > M - 1) gr = M - 1;
                const float* src = Ap + (size_t)gr * lda + k0 + col;
                const float4 v = *(const float4*)src;
                float* d = &As[row * (KB + APAD) + col];
                d[0] = v.x; d[1] = v.y; d[2] = v.z; d[3] = v.w;
                if (k0 + KB < K)   // prefetch next k-slice of the streamed operand
                    __builtin_prefetch(src + KB, 0, 1);
            }
            // ---- stage B tile (one float4 per thread) ----
            if (TRANSB) {
                // W tile: 64 rows (n) x 16 cols (k)
                const int row = tid >> 2;
                const int col = (tid & 3) * 4;
                const float4 v = *(const float4*)(Bp + (size_t)(n0 + row) * ldb + k0 + col);
                float* d = &Bs[row * (KB + APAD) + col];
                d[0] = v.x; d[1] = v.y; d[2] = v.z; d[3] = v.w;
            } else {
                // B tile: 16 rows (k) x 64 cols (n)
                const int row = tid >> 4;
                const int col = (tid & 15) * 4;
                const float4 v = *(const float4*)(Bp + (size_t)(k0 + row) * ldb + n0 + col);
                float* d = &Bs[row * (64 + APAD) + col];
                d[0] = v.x; d[1] = v.y; d[2] = v.z; d[3] = v.w;
            }
            __syncthreads();

            // ---- 4 WMMA k-steps from LDS ----
#pragma unroll
            for (int kk = 0; kk < KB; kk += 4) {
                const int krow = kk + 2 * half;                 // even -> 8B aligned
                const v2f a = *(const v2f*)&As[(wave * 16 + r) * (KB + APAD) + krow];
#pragma unroll
                for (int t = 0; t < 4; ++t) {
                    v2f b;
                    if (TRANSB) {
                        b = *(const v2f*)&Bs[(t * 16 + r) * (KB + APAD) + krow];
                    } else {
                        b.x = Bs[krow       * (64 + APAD) + t * 16 + r];
                        b.y = Bs[(krow + 1) * (64 + APAD) + t * 16 + r];
                    }
                    acc[t] = __builtin_amdgcn_wmma_f32_16x16x4_f32(
                        false, a, false, b, (short)0, acc[t], false, false);
                }
            }
        }
    }

    // ---- epilogue ----
    const int row_base = m0 + wave * 16 + 8 * half;
#pragma unroll
    for (int t = 0; t < 4; ++t) {
        const int n = n0 + t * 16 + r;
        const float bv = bias ? bias[n] : 0.0f;
#pragma unroll
        for (int j = 0; j < 8; ++j) {
            const int row = row_base + j;
            if (row < M) {
                float v = acc[t][j] + bv;
                if (RELU) v = fmaxf(v, 0.0f);
                C[(size_t)row * ldc + n] = v;
            }
        }
    }
}

// ---------------------------------------------------------------------------
// One-wave WMMA GEMM (kept for the tiny N=32 classifier layer).
//   C = A[M,K] @ W[N,K]^T + bias, relu optional.  Exact 16-tiles assumed.
// ---------------------------------------------------------------------------
template <bool RELU>
__global__ __launch_bounds__(32) void wmma_gemm_f32_small(
    const float* __restrict__ A, int lda,
    const float* __restrict__ B, int ldb,
    const float* __restrict__ bias,
    float* __restrict__ C, int ldc,
    int K)
{
    const int lane = threadIdx.x & 31;
    const int half = lane >> 4;
    const int r    = lane & 15;
    const int m0   = blockIdx.x * 16;
    const int n0   = blockIdx.y * 16;

    v8f acc = (v8f){0.f,0.f,0.f,0.f,0.f,0.f,0.f,0.f};
    const float* arow = A + (size_t)(m0 + r) * lda + 2 * half;
    const float* brow = B + (size_t)(n0 + r) * ldb + 2 * half;
    for (int k = 0; k < K; k += 4) {
        const v2f a = *(const v2f*)(arow + k);
        const v2f b = *(const v2f*)(brow + k);
        acc = __builtin_amdgcn_wmma_f32_16x16x4_f32(
            false, a, false, b, (short)0, acc, false, false);
    }
    const int n = n0 + r;
    const float bv = bias ? bias[n] : 0.0f;
#pragma unroll
    for (int j = 0; j < 8; ++j) {
        float v = acc[j] + bv;
        if (RELU) v = fmaxf(v, 0.0f);
        C[(size_t)(m0 + 8 * half + j) * ldc + n] = v;
    }
}

// --------------------------- scatter-mean ----------------------------------
__global__ void scatter_add_kernel(const float* __restrict__ feat, int F,
                                   const int* __restrict__ src,
                                   const int* __restrict__ dst, int E,
                                   float* __restrict__ agg,
                                   float* __restrict__ cnt)
{
    const int e = blockIdx.x;
    if (e >= E) return;
    const int s = src[e];
    const int d = dst[e];
    const float* xs = feat + (size_t)s * F;
    float* ag = agg + (size_t)d * F;
    for (int i = threadIdx.x; i < F; i += blockDim.x)
        atomicAdd(&ag[i], xs[i]);
    if (threadIdx.x == 0) atomicAdd(&cnt[d], 1.0f);
}

__global__ void div_cnt_kernel(float* __restrict__ agg,
                               const float* __restrict__ cnt,
                               int N, int F)
{
    const size_t idx = (size_t)blockIdx.x * blockDim.x + threadIdx.x;
    if (idx >= (size_t)N * F) return;
    const int row = (int)(idx / F);
    agg[idx] = agg[idx] / fmaxf(cnt[row], 1.0f);
}

// --------------------------- row-wise ops ----------------------------------
__global__ __launch_bounds__(32) void log_softmax_rows(float* __restrict__ h,
                                                       int N, int F)
{
    const int row = blockIdx.x;
    if (row >= N) return;
    float* p = h + (size_t)row * F;
    const int lane = threadIdx.x;
    float m = -3.402823e38f;
    for (int i = lane; i < F; i += 32) m = fmaxf(m, p[i]);
    m = wave_max(m);
    float s = 0.0f;
    for (int i = lane; i < F; i += 32) s += expf(p[i] - m);
    s = wave_sum(s);
    const float ls = logf(s);
    for (int i = lane; i < F; i += 32) p[i] = p[i] - m - ls;
}

__global__ __launch_bounds__(32) void l2n_rows(float* __restrict__ v, int N, int F)
{
    const int row = blockIdx.x;
    if (row >= N) return;
    float* p = v + (size_t)row * F;
    const int lane = threadIdx.x;
    float s = 0.0f;
    for (int i = lane; i < F; i += 32) { const float x = p[i]; s += x * x; }
    s = wave_sum(s);
    const float inv = 1.0f / fmaxf(sqrtf(s), EPS);
    for (int i = lane; i < F; i += 32) p[i] *= inv;
}

__global__ __launch_bounds__(32) void div_loss_rows(const float* __restrict__ t1,
                                                    const float* __restrict__ t2,
                                                    float* __restrict__ loss,
                                                    int N, int F)
{
    const int row = blockIdx.x;
    if (row >= N) return;
    const float* a = t1 + (size_t)row * F;
    const float* b = t2 + (size_t)row * F;
    const int lane = threadIdx.x;
    float s1 = 0.0f, s2 = 0.0f;
    for (int i = lane; i < F; i += 32) {
        const float x = a[i]; s1 += x * x;
        const float y = b[i]; s2 += y * y;
    }
    s1 = wave_sum(s1);
    s2 = wave_sum(s2);
    const float i1 = 1.0f / fmaxf(sqrtf(s1), EPS);
    const float i2 = 1.0f / fmaxf(sqrtf(s2), EPS);
    float acc = 0.0f;
    for (int i = lane; i < F; i += 32) {
        const float d = a[i] * i1 - b[i] * i2;
        acc += d * d;
    }
    acc = wave_sum(acc);
    if (lane == 0) atomicAdd(loss, acc / (float)N);
}

// final classifier layer: l[row] = dot(c1[row, 0:32], Wc2) + bc2
__global__ __launch_bounds__(32) void clf_out_kernel(const float* __restrict__ c1,
                                                     const float* __restrict__ Wc2,
                                                     const float* __restrict__ bc2,
                                                     float* __restrict__ l, int N)
{
    const int row = blockIdx.x;
    if (row >= N) return;
    const int lane = threadIdx.x;
    float v = c1[(size_t)row * 32 + lane] * Wc2[lane];
    v = wave_sum(v);
    if (lane == 0) l[row] = v + bc2[0];
}

__global__ void finalize_kernel(const float* __restrict__ l1,
                                const float* __restrict__ l2,
                                const float* __restrict__ loss,
                                float* __restrict__ out, int n2)
{
    const int i = blockIdx.x * blockDim.x + threadIdx.x;
    if (i < n2) {
        out[2 * i + 0] = l1[i];
        out[2 * i + 1] = l2[i];
    }
    if (i == 0) out[2 * n2] = loss[0];
}

// ---------------------------------------------------------------------------
extern "C" void kernel_launch(void* const* d_in, const int* in_sizes, int n_in,
                              void* d_out, int out_size, void* d_ws, size_t ws_size,
                              hipStream_t stream)
{
    // setup_inputs() dict order:
    const float* x    = (const float*)d_in[0];   // [40000, 512]
    const float* ppr  = (const float*)d_in[1];   // [8000, 8000]
    const int*   src0 = (const int*)d_in[2];
    const int*   dst0 = (const int*)d_in[3];
    const int*   src1 = (const int*)d_in[4];
    const int*   dst1 = (const int*)d_in[5];
    // d_in[6]=n1, d_in[7]=n2 (scalars, known statically)
    const float* Wl0 = (const float*)d_in[8];
    const float* bl0 = (const float*)d_in[9];
    const float* Wr0 = (const float*)d_in[10];
    const float* Wl1 = (const float*)d_in[11];
    const float* bl1 = (const float*)d_in[12];
    const float* Wr1 = (const float*)d_in[13];
    const float* Wa  = (const float*)d_in[14];
    const float* ba  = (const float*)d_in[15];
    const float* Wb  = (const float*)d_in[16];
    const float* bb  = (const float*)d_in[17];
    const float* W1  = (const float*)d_in[18];
    const float* W2  = (const float*)d_in[19];
    const float* Wp1 = (const float*)d_in[20];
    const float* bp1 = (const float*)d_in[21];
    const float* Wp2 = (const float*)d_in[22];
    const float* bp2 = (const float*)d_in[23];
    const float* Wc1 = (const float*)d_in[24];
    const float* bc1 = (const float*)d_in[25];
    const float* Wc2 = (const float*)d_in[26];
    const float* bc2 = (const float*)d_in[27];

    const int N1 = 8000, N2 = 2000, IN = 512, HID = 256;
    const int E0 = in_sizes[2];
    const int E1 = in_sizes[4];

    float* ws = (float*)d_ws;
    // ---- workspace layout (floats), with dead-buffer reuse ----
    float* agg0 = ws;                       // 8000*512
    float* cnt0 = agg0 + 4096000;           // 8000
    float* h1   = cnt0 + 8000;              // 8000*256
    float* agg1 = h1 + 2048000;             // 2000*256
    float* cnt1 = agg1 + 512000;            // 2000
    float* ebds = cnt1 + 2000;              // 2000*256  (h2 -> log_softmax in place)
    float* t1   = ebds + 512000;            // 2000*256
    float* t2   = t1 + 512000;              // 2000*256
    float* loss = t2 + 512000;              // 1 (padded 16)
    float* xp1  = loss + 16;                // 8000*512
    float* xp3  = xp1 + 4096000;            // 2000*512
    float* xp4  = xp3 + 1024000;            // 2000*256
    // reuse of dead buffers:
    float* xp2  = agg0;                     // 8000*512 (agg0 dead after h1)
    float* phid = agg1;                     // 2000*256 (agg1 dead after h2)
    float* pout = t1;                       // 2000*256 (t1 dead after div_loss)
    float* c1b  = t2;                       // 2000*32  (t2 dead after div_loss)
    float* l1v  = cnt0;                     // 2000     (cnt0 dead after layer0 mean)
    float* l2v  = cnt0 + 2000;              // 2000

    float* out = (float*)d_out;

    const dim3 gBig(63, 8);    // ceil(8000/128) x 512/64
    const dim3 gBigH(63, 4);   // ceil(8000/128) x 256/64
    const dim3 gSm512(16, 8);  // ceil(2000/128) x 512/64
    const dim3 gSm256(16, 4);  // ceil(2000/128) x 256/64

    // ---- zero accumulators ----
    hipMemsetAsync(agg0, 0, (size_t)4096000 * sizeof(float), stream);
    hipMemsetAsync(cnt0, 0, (size_t)8000 * sizeof(float), stream);
    hipMemsetAsync(agg1, 0, (size_t)512000 * sizeof(float), stream);
    hipMemsetAsync(cnt1, 0, (size_t)2000 * sizeof(float), stream);
    hipMemsetAsync(loss, 0, 16 * sizeof(float), stream);

    // ---- SAGE layer 0: scatter-mean then fused dual GEMM + ReLU ----
    scatter_add_kernel<<<E0, 256, 0, stream>>>(x, IN, src0, dst0, E0, agg0, cnt0);
    div_cnt_kernel<<<(N1 * IN + 255) / 256, 256, 0, stream>>>(agg0, cnt0, N1, IN);
    // h1 = relu(agg0 @ Wl0^T + x[:N1] @ Wr0^T + bl0)   [8000, 256]
    wmma_gemm_f32_tiled<true, true, true><<<gBigH, 256, 0, stream>>>(
        agg0, IN, Wl0, IN, x, Wr0, bl0, h1, HID, N1, IN);

    // ---- SAGE layer 1 ----
    scatter_add_kernel<<<E1, 256, 0, stream>>>(h1, HID, src1, dst1, E1, agg1, cnt1);
    div_cnt_kernel<<<(N2 * HID + 255) / 256, 256, 0, stream>>>(agg1, cnt1, N2, HID);
    // h2 = agg1 @ Wl1^T + h1[:N2] @ Wr1^T + bl1   [2000, 256]
    wmma_gemm_f32_tiled<true, true, false><<<gSm256, 256, 0, stream>>>(
        agg1, HID, Wl1, HID, h1, Wr1, bl1, ebds, HID, N2, HID);

    // ---- ebds = log_softmax(h2) ----
    log_softmax_rows<<<N2, 32, 0, stream>>>(ebds, N2, HID);

    // ---- divergence loss ----
    wmma_gemm_f32_tiled<true, false, false><<<gSm256, 256, 0, stream>>>(
        ebds, HID, W1, HID, nullptr, nullptr, nullptr, t1, HID, N2, HID);
    wmma_gemm_f32_tiled<true, false, false><<<gSm256, 256, 0, stream>>>(
        ebds, HID, W2, HID, nullptr, nullptr, nullptr, t2, HID, N2, HID);
    div_loss_rows<<<N2, 32, 0, stream>>>(t1, t2, loss, N2, HID);

    // ---- PPR branch ----
    // xp1 = ppr[:8000,:8000] @ x[:8000]     [8000, 512]  (big NN GEMM)
    wmma_gemm_f32_tiled<false, false, false><<<gBig, 256, 0, stream>>>(
        ppr, N1, x, IN, nullptr, nullptr, nullptr, xp1, IN, N1, N1);
    // xp2 = relu(xp1 @ Wa^T + ba)           [8000, 512]
    wmma_gemm_f32_tiled<true, false, true><<<gBig, 256, 0, stream>>>(
        xp1, IN, Wa, IN, nullptr, nullptr, ba, xp2, IN, N1, IN);
    // xp3 = ppr[:2000,:2000] @ xp2[:2000]   [2000, 512]  (lda = 8000!)
    wmma_gemm_f32_tiled<false, false, false><<<gSm512, 256, 0, stream>>>(
        ppr, N1, xp2, IN, nullptr, nullptr, nullptr, xp3, IN, N2, N2);
    // xp4 = xp3 @ Wb^T + bb                 [2000, 256]
    wmma_gemm_f32_tiled<true, false, false><<<gSm256, 256, 0, stream>>>(
        xp3, IN, Wb, IN, nullptr, nullptr, bb, xp4, HID, N2, IN);

    // ---- g1 = l2n(ebds) in place; g2 = l2n(xp4) in place ----
    l2n_rows<<<N2, 32, 0, stream>>>(ebds, N2, HID);
    l2n_rows<<<N2, 32, 0, stream>>>(xp4, N2, HID);

    // ---- branch 1: clf(proj(g1)) -> l1v ----
    wmma_gemm_f32_tiled<true, false, true><<<gSm256, 256, 0, stream>>>(
        ebds, HID, Wp1, HID, nullptr, nullptr, bp1, phid, HID, N2, HID);
    wmma_gemm_f32_tiled<true, false, false><<<gSm256, 256, 0, stream>>>(
        phid, HID, Wp2, HID, nullptr, nullptr, bp2, pout, HID, N2, HID);
    wmma_gemm_f32_small<true><<<dim3(N2 / 16, 2), 32, 0, stream>>>(
        pout, HID, Wc1, HID, bc1, c1b, 32, HID);
    clf_out_kernel<<<N2, 32, 0, stream>>>(c1b, Wc2, bc2, l1v, N2);

    // ---- branch 2: clf(proj(g2)) -> l2v ----
    wmma_gemm_f32_tiled<true, false, true><<<gSm256, 256, 0, stream>>>(
        xp4, HID, Wp1, HID, nullptr, nullptr, bp1, phid, HID, N2, HID);
    wmma_gemm_f32_tiled<true, false, false><<<gSm256, 256, 0, stream>>>(
        phid, HID, Wp2, HID, nullptr, nullptr, bp2, pout, HID, N2, HID);
    wmma_gemm_f32_small<true><<<dim3(N2 / 16, 2), 32, 0, stream>>>(
        pout, HID, Wc1, HID, bc1, c1b, 32, HID);
    clf_out_kernel<<<N2, 32, 0, stream>>>(c1b, Wc2, bc2, l2v, N2);

    // ---- write [2000,2] logits + div_loss ----
    finalize_kernel<<<(N2 + 255) / 256, 256, 0, stream>>>(l1v, l2v, loss, out, N2);
}